// VinaFreeEnergy_29746943492206
// MI455X (gfx1250) — compile-verified
//
#include <hip/hip_runtime.h>
#include <math.h>

// ---------------------------------------------------------------------------
// VinaFreeEnergy for MI455X (gfx1250, wave32).
// Latency-bound neighbor-list pipeline; the 512x8192 cell-atom distance
// matrix is computed as a fused V_WMMA_F32_16X16X4_F32 (d = |a|^2 - 2a.c + |c|^2)
// and reduced to per-cell top-8 entirely in registers (sortable u64 keys:
// float-bits<<32 | index  ==> exact stable tie-break like jax.lax.top_k).
// ---------------------------------------------------------------------------

#define N_ATOMS     8192
#define M_NBRS      8
#define N_CELLS     512
#define N_NBR_CELLS 27
#define L_CAND      216     // 27 * 8
#define PAD_C       32.0f   // 2 * STOP

typedef float v2f __attribute__((ext_vector_type(2)));
typedef float v8f __attribute__((ext_vector_type(8)));

// cells = transpose(stack(meshgrid(r,r,r))).reshape(512,3), r = 0,2,...,14
// flat index c -> ( 2*((c>>3)&7), 2*(c&7), 2*((c>>6)&7) )
__device__ __forceinline__ void cell_coord(int c, float& x, float& y, float& z) {
  x = 2.0f * (float)((c >> 3) & 7);
  y = 2.0f * (float)(c & 7);
  z = 2.0f * (float)((c >> 6) & 7);
}

// insert k into ascending sorted key[8] (keep 8 smallest)
__device__ __forceinline__ void insert8(unsigned long long k, unsigned long long key[8]) {
  if (k < key[7]) {
    key[7] = k;
#pragma unroll
    for (int s = 7; s >= 1; --s) {
      unsigned long long a = key[s - 1], b = key[s];
      bool sw = (b < a);
      key[s - 1] = sw ? b : a;
      key[s]     = sw ? a : b;
    }
  }
}

#define CE(i, j)                                            \
  do {                                                      \
    unsigned long long _a = s[i], _b = s[j];                \
    bool _sw = (_b < _a);                                   \
    s[i] = _sw ? _b : _a;                                   \
    s[j] = _sw ? _a : _b;                                   \
  } while (0)

// merge two ascending sorted 8-lists, keep the 8 smallest (sorted) in key[].
// Bitonic: concat(a, reverse(b)) is bitonic; first CE stage selects the 8
// smallest; then sort the (bitonic) low half.
__device__ __forceinline__ void merge8(unsigned long long key[8],
                                       const unsigned long long other[8]) {
  unsigned long long s[16];
#pragma unroll
  for (int i = 0; i < 8; ++i) { s[i] = key[i]; s[8 + i] = other[7 - i]; }
#pragma unroll
  for (int i = 0; i < 8; ++i) CE(i, i + 8);
  CE(0, 4); CE(1, 5); CE(2, 6); CE(3, 7);
  CE(0, 2); CE(1, 3); CE(4, 6); CE(5, 7);
  CE(0, 1); CE(2, 3); CE(4, 5); CE(6, 7);
#pragma unroll
  for (int i = 0; i < 8; ++i) key[i] = s[i];
}

__device__ __forceinline__ unsigned long long shfl_xor_u64(unsigned long long v, int mask) {
  unsigned int lo = (unsigned int)v;
  unsigned int hi = (unsigned int)(v >> 32);
  lo = __shfl_xor(lo, mask, 32);
  hi = __shfl_xor(hi, mask, 32);
  return ((unsigned long long)hi << 32) | (unsigned long long)lo;
}

// ---------------------------------------------------------------------------
// K0: per-atom squared norms (feeds the WMMA bias trick)
// ---------------------------------------------------------------------------
__global__ void k_atom_sq(const float* __restrict__ X, float* __restrict__ atom_sq) {
  int i = blockIdx.x * blockDim.x + threadIdx.x;
  if (i < N_ATOMS) {
    float x = X[3 * i], y = X[3 * i + 1], z = X[3 * i + 2];
    atom_sq[i] = x * x + y * y + z * z;
  }
}

// ---------------------------------------------------------------------------
// K1: per-cell top-8 closest atoms via fused WMMA distance tiles.
// One wave per 16-cell tile. D = A(atoms 16x4) * B(cells 4x16) + C.
//   A row m: (-2ax, -2ay, -2az, |a|^2)   B col n: (cx, cy, cz, 1)   C: |c|^2
// D layout: lanes 0-15 -> cell col = lane, atom rows 0-7 (vgpr v);
//           lanes 16-31 -> cell col = lane-16, atom rows 8-15.
// Each lane keeps a top-8 for its cell; lane n merges with lane n+16 at the end.
// ---------------------------------------------------------------------------
__global__ void __launch_bounds__(32) k_cell_top8(const float* __restrict__ X,
                                                  const float* __restrict__ atom_sq,
                                                  int* __restrict__ closest_atoms) {
  const int  lane = threadIdx.x;
  const int  n    = lane & 15;
  const bool hi   = lane >= 16;
  const int  cellId = blockIdx.x * 16 + n;

  float cx, cy, cz;
  cell_coord(cellId, cx, cy, cz);
  const float csq = cx * cx + cy * cy + cz * cz;

  v2f B;                               // 4x16 B: lanes 0-15 hold K0,K1; 16-31 hold K2,K3
  B.x = hi ? cz : cx;
  B.y = hi ? 1.0f : cy;
  v8f C;
#pragma unroll
  for (int v = 0; v < 8; ++v) C[v] = csq;

  unsigned long long key[8];
#pragma unroll
  for (int v = 0; v < 8; ++v) key[v] = ~0ull;

  const int rofs = hi ? 8 : 0;
  for (int t = 0; t < N_ATOMS / 16; ++t) {
    const int abase = t * 16;
    const int ai    = abase + n;
    const float ax = X[3 * ai], ay = X[3 * ai + 1], az = X[3 * ai + 2];
    const float asq = atom_sq[ai];
    v2f A;                             // 16x4 A: lanes 0-15 hold K0,K1; 16-31 hold K2,K3
    A.x = hi ? (-2.0f * az) : (-2.0f * ax);
    A.y = hi ? asq : (-2.0f * ay);

    v8f D = __builtin_amdgcn_wmma_f32_16x16x4_f32(false, A, false, B,
                                                  (short)0, C, false, false);
#pragma unroll
    for (int v = 0; v < 8; ++v) {
      float d = D[v];
      if (d < 0.0f) d = 0.0f;  // keep float bits monotone-sortable (expansion rounding)
      unsigned long long k =
          ((unsigned long long)__float_as_uint(d) << 32) | (unsigned int)(abase + rofs + v);
      insert8(k, key);
    }
  }

  // merge atom-rows 0-7 list (lane n) with rows 8-15 list (lane n+16)
  unsigned long long other[8];
#pragma unroll
  for (int v = 0; v < 8; ++v) other[v] = shfl_xor_u64(key[v], 16);
  merge8(key, other);

  if (!hi) {
#pragma unroll
    for (int m = 0; m < 8; ++m)
      closest_atoms[cellId * M_NBRS + m] = (int)(unsigned int)(key[m] & 0xFFFFFFFFull);
  }
}

// ---------------------------------------------------------------------------
// K2: nearest cell per atom (argmin, first-wins like jnp.argmin)
// ---------------------------------------------------------------------------
__global__ void k_atom_cell(const float* __restrict__ X, int* __restrict__ cells_for_atoms) {
  int i = blockIdx.x * blockDim.x + threadIdx.x;
  if (i >= N_ATOMS) return;
  float x = X[3 * i], y = X[3 * i + 1], z = X[3 * i + 2];
  float best = 3.4e38f;
  int   bc   = 0;
  for (int c = 0; c < N_CELLS; ++c) {
    float cx, cy, cz;
    cell_coord(c, cx, cy, cz);
    float dx = x - cx, dy = y - cy, dz = z - cz;
    float d = dx * dx + dy * dy + dz * dz;
    if (d < best) { best = d; bc = c; }
  }
  cells_for_atoms[i] = bc;
}

// ---------------------------------------------------------------------------
// K3: 27 nearest cells per cell (stable top-27, sorted ascending)
// ---------------------------------------------------------------------------
__global__ void k_cell_top27(int* __restrict__ nbr_cells) {
  int c = blockIdx.x * blockDim.x + threadIdx.x;
  if (c >= N_CELLS) return;
  float x, y, z;
  cell_coord(c, x, y, z);
  unsigned long long key[N_NBR_CELLS];
  for (int t = 0; t < N_NBR_CELLS; ++t) key[t] = ~0ull;
  for (int oc = 0; oc < N_CELLS; ++oc) {
    float cx, cy, cz;
    cell_coord(oc, cx, cy, cz);
    float dx = x - cx, dy = y - cy, dz = z - cz;
    float d = dx * dx + dy * dy + dz * dz;
    unsigned long long k = ((unsigned long long)__float_as_uint(d) << 32) | (unsigned int)oc;
    if (k < key[N_NBR_CELLS - 1]) {
      key[N_NBR_CELLS - 1] = k;
      for (int s = N_NBR_CELLS - 1; s >= 1; --s) {
        if (key[s] < key[s - 1]) {
          unsigned long long tmp = key[s]; key[s] = key[s - 1]; key[s - 1] = tmp;
        }
      }
    }
  }
  for (int t = 0; t < N_NBR_CELLS; ++t)
    nbr_cells[c * N_NBR_CELLS + t] = (int)(unsigned int)(key[t] & 0xFFFFFFFFull);
}

// ---------------------------------------------------------------------------
// K4: per-atom candidate list (216), dedup-first-occurrence, drop == nbrs[0],
// distances (pad coord for invalid), stable top-8 -> neighbor indices / -1.
// One wave per atom; 8 atoms per 256-thread block. Hypercube shuffle merge.
// ---------------------------------------------------------------------------
__global__ void __launch_bounds__(256) k_build_nbrs(const float* __restrict__ X,
                                                    const int* __restrict__ cells_for_atoms,
                                                    const int* __restrict__ nbr_cells,
                                                    const int* __restrict__ closest_atoms,
                                                    int* __restrict__ nbr_list) {
  __shared__ int           s_nbrs[8][L_CAND];
  __shared__ unsigned char s_valid[8][L_CAND];
  const int wave = threadIdx.x >> 5;
  const int lane = threadIdx.x & 31;
  const int atom = blockIdx.x * 8 + wave;

  const int c = cells_for_atoms[atom];
  for (int l = lane; l < L_CAND; l += 32) {
    int t = l >> 3, m = l & 7;
    int nc = nbr_cells[c * N_NBR_CELLS + t];
    s_nbrs[wave][l] = closest_atoms[nc * M_NBRS + m];
  }
  __syncthreads();

  const int   first = s_nbrs[wave][0];
  const float ax = X[3 * atom], ay = X[3 * atom + 1], az = X[3 * atom + 2];
  const float pdx = ax - PAD_C, pdy = ay - PAD_C, pdz = az - PAD_C;
  const float padDist = pdx * pdx + pdy * pdy + pdz * pdz;

  unsigned long long key[8];
#pragma unroll
  for (int v = 0; v < 8; ++v) key[v] = ~0ull;

  for (int l = lane; l < L_CAND; l += 32) {
    const int idx = s_nbrs[wave][l];
    bool seen = false;
    for (int p = 0; p < l; ++p)
      if (s_nbrs[wave][p] == idx) { seen = true; break; }
    const bool valid = (!seen) && (idx != first);
    s_valid[wave][l] = valid ? 1 : 0;
    float d;
    if (valid) {
      float dx = ax - X[3 * idx], dy = ay - X[3 * idx + 1], dz = az - X[3 * idx + 2];
      d = dx * dx + dy * dy + dz * dz;
    } else {
      d = padDist;
    }
    unsigned long long k = ((unsigned long long)__float_as_uint(d) << 32) | (unsigned int)l;
    insert8(k, key);
  }
  __syncthreads();

  // hypercube merge: after 5 rounds every lane holds the wave-global top-8
#pragma unroll
  for (int off = 16; off >= 1; off >>= 1) {
    unsigned long long other[8];
#pragma unroll
    for (int v = 0; v < 8; ++v) other[v] = shfl_xor_u64(key[v], off);
    merge8(key, other);
  }

  if (lane < M_NBRS) {
    int l = (int)(unsigned int)(key[lane] & 0xFFFFFFFFull);
    nbr_list[atom * M_NBRS + lane] = s_valid[wave][l] ? s_nbrs[wave][l] : -1;
  }
}

// ---------------------------------------------------------------------------
// K5/K6: pair energies + deterministic two-stage tree reduction
// ---------------------------------------------------------------------------
__global__ void __launch_bounds__(256) k_energy(const float* __restrict__ X,
                                                const int* __restrict__ nbr_list,
                                                const float* __restrict__ cw,
                                                const float* __restrict__ w,
                                                float* __restrict__ partials) {
  __shared__ float red[256];
  const int gid = blockIdx.x * 256 + threadIdx.x;
  const int n = gid >> 3, m = gid & 7;

  int j  = nbr_list[n * M_NBRS + m];
  int jj = (j < 0) ? (N_ATOMS - 1) : j;  // -1 padding wraps to last atom, like X[-1]
  float dx = X[3 * n] - X[3 * jj];
  float dy = X[3 * n + 1] - X[3 * jj + 1];
  float dz = X[3 * n + 2] - X[3 * jj + 2];
  float d = dx * dx + dy * dy + dz * dz;  // SQUARED distance, faithful to source

  float rep = (d < 0.0f) ? d * d : 0.0f;
  float hyd = (d < 0.5f) ? 1.0f : ((d < 1.5f) ? (1.5f - d) : 0.0f);
  float hb  = (d < -0.7f) ? 1.0f : ((d < 0.0f) ? (-d / 0.7f) : 0.0f);
  float t1 = d * 2.0f;            // d / 0.5
  float g1 = expf(-t1 * t1);
  float t2 = (d - 3.0f) * 0.5f;   // (d - 3) / 2
  float g2 = expf(-t2 * t2);
  float inter = cw[0] * rep + cw[1] * hyd + cw[2] * hb + cw[3] * g1 + cw[4] * g2;
  float th = (d < 8.0f) ? inter : 0.0f;
  float e  = th / (1.0f + w[0]);  // NROT = 1.0

  red[threadIdx.x] = e;
  __syncthreads();
  for (int s = 128; s > 0; s >>= 1) {
    if (threadIdx.x < s) red[threadIdx.x] += red[threadIdx.x + s];
    __syncthreads();
  }
  if (threadIdx.x == 0) partials[blockIdx.x] = red[0];
}

__global__ void __launch_bounds__(256) k_finalize(const float* __restrict__ partials,
                                                  float* __restrict__ out) {
  __shared__ float red[256];
  red[threadIdx.x] = partials[threadIdx.x];
  __syncthreads();
  for (int s = 128; s > 0; s >>= 1) {
    if (threadIdx.x < s) red[threadIdx.x] += red[threadIdx.x + s];
    __syncthreads();
  }
  if (threadIdx.x == 0) out[0] = red[0];
}

// ---------------------------------------------------------------------------
// host-side launch
// ---------------------------------------------------------------------------
extern "C" void kernel_launch(void* const* d_in, const int* in_sizes, int n_in,
                              void* d_out, int out_size, void* d_ws, size_t ws_size,
                              hipStream_t stream) {
  (void)in_sizes; (void)n_in; (void)out_size; (void)ws_size;

  const float* X     = (const float*)d_in[0];
  // d_in[1] = Z (atom types): accepted but unused by the reference math
  const float* combo = (const float*)d_in[2];
  const float* w     = (const float*)d_in[3];
  float*       out   = (float*)d_out;

  // workspace layout (~392 KB total)
  char*  ws              = (char*)d_ws;
  float* atom_sq         = (float*)(ws + 0);                 //  32 KB
  int*   closest_atoms   = (int*)(ws + 32768);               //  16 KB (512*8)
  int*   cells_for_atoms = (int*)(ws + 49152);               //  32 KB (8192)
  int*   nbr_cells       = (int*)(ws + 81920);               //  54 KB (512*27)
  int*   nbr_list        = (int*)(ws + 137216);              // 256 KB (8192*8)
  float* partials        = (float*)(ws + 399360);            //   1 KB (256)

  k_atom_sq   <<<N_ATOMS / 256, 256, 0, stream>>>(X, atom_sq);
  k_cell_top8 <<<N_CELLS / 16,   32, 0, stream>>>(X, atom_sq, closest_atoms);
  k_atom_cell <<<N_ATOMS / 256, 256, 0, stream>>>(X, cells_for_atoms);
  k_cell_top27<<<N_CELLS / 256, 256, 0, stream>>>(nbr_cells);
  k_build_nbrs<<<N_ATOMS / 8,   256, 0, stream>>>(X, cells_for_atoms, nbr_cells,
                                                  closest_atoms, nbr_list);
  k_energy    <<<(N_ATOMS * M_NBRS) / 256, 256, 0, stream>>>(X, nbr_list, combo, w, partials);
  k_finalize  <<<1, 256, 0, stream>>>(partials, out);
}